// CPENetwork_ResAG_19335942766952
// MI455X (gfx1250) — compile-verified
//
#include <hip/hip_runtime.h>
#include <hip/hip_bf16.h>

// ---- problem constants (from reference) ----
#define BB 128
#define TT 77
#define DD 768
#define NN 16
#define HH 16
#define RR 4
#define OO 320
#define MM (BB * TT)          // 9856 rows = 154 * 64 exactly
#define MBLK 64               // rows per wave tile (4 x 16 sub-tiles)
#define MTILES (MM / MBLK)    // 154
#define OTILES (OO / 16)      // 20

typedef __attribute__((ext_vector_type(2))) float v2f;
typedef __attribute__((ext_vector_type(8))) float v8f;

// =====================================================================
// K1: wq[b,n,t,h] = sum_d x[b,t,d] * w_qs[n,d,h]
// One wave per (64-row mtile, concept n), 4 accumulators, explicit
// ping-pong double buffering of the 5 fragment loads per k-step so the
// WMMAs for set k overlap the loads for set k+4.
// =====================================================================
__global__ void __attribute__((amdgpu_waves_per_eu(1)))
k1_wq_wmma(const float* __restrict__ X,
           const float* __restrict__ Wq,
           float* __restrict__ wq) {
  const int lane = threadIdx.x & 31;
  const int wave = threadIdx.x >> 5;
  const int tile = blockIdx.x * 4 + wave;     // MTILES*NN = 2464 tiles exactly
  const int n  = tile & (NN - 1);
  const int tm = tile >> 4;                   // 0..153
  const int lrow = lane & 15;
  const int kseg = lane >> 4;                 // 0/1 -> K offset 0/2
  const int h = lane & 15;

  // fragment base pointers (per-lane), inner loop indexes by k only
  const float* xa0 = X + (size_t)(tm * MBLK +  0 + lrow) * DD + 2 * kseg;
  const float* xa1 = X + (size_t)(tm * MBLK + 16 + lrow) * DD + 2 * kseg;
  const float* xa2 = X + (size_t)(tm * MBLK + 32 + lrow) * DD + 2 * kseg;
  const float* xa3 = X + (size_t)(tm * MBLK + 48 + lrow) * DD + 2 * kseg;
  const float* wbp = Wq + (size_t)n * DD * HH + (size_t)(2 * kseg) * HH + h;

  v8f acc0 = {}, acc1 = {}, acc2 = {}, acc3 = {};

  v2f bA, a0A, a1A, a2A, a3A;   // set A (even k-step)
  v2f bB, a0B, a1B, a2B, a3B;   // set B (odd k-step)

#define LD_SET(kk, bv, f0, f1, f2, f3)                   \
  bv.x = wbp[(size_t)(kk) * HH];                         \
  bv.y = wbp[(size_t)(kk) * HH + HH];                    \
  f0 = *(const v2f*)(xa0 + (kk));                        \
  f1 = *(const v2f*)(xa1 + (kk));                        \
  f2 = *(const v2f*)(xa2 + (kk));                        \
  f3 = *(const v2f*)(xa3 + (kk));

#define WMMA_SET(bv, f0, f1, f2, f3)                                                              \
  acc0 = __builtin_amdgcn_wmma_f32_16x16x4_f32(false, f0, false, bv, (short)0, acc0, false, false); \
  acc1 = __builtin_amdgcn_wmma_f32_16x16x4_f32(false, f1, false, bv, (short)0, acc1, false, false); \
  acc2 = __builtin_amdgcn_wmma_f32_16x16x4_f32(false, f2, false, bv, (short)0, acc2, false, false); \
  acc3 = __builtin_amdgcn_wmma_f32_16x16x4_f32(false, f3, false, bv, (short)0, acc3, false, false);

  LD_SET(0, bA, a0A, a1A, a2A, a3A)
  int k = 0;
  for (; k < DD - 8; k += 8) {
    LD_SET(k + 4, bB, a0B, a1B, a2B, a3B)
    WMMA_SET(bA, a0A, a1A, a2A, a3A)
    LD_SET(k + 8, bA, a0A, a1A, a2A, a3A)
    WMMA_SET(bB, a0B, a1B, a2B, a3B)
  }
  // tail: k == DD-8; set A holds k, load k+4 into set B
  LD_SET(k + 4, bB, a0B, a1B, a2B, a3B)
  WMMA_SET(bA, a0A, a1A, a2A, a3A)
  WMMA_SET(bB, a0B, a1B, a2B, a3B)

  v8f accs[4] = {acc0, acc1, acc2, acc3};
#pragma unroll
  for (int sub = 0; sub < 4; ++sub) {
    const int mbase = tm * MBLK + sub * 16 + (kseg ? 8 : 0);
#pragma unroll
    for (int i = 0; i < 8; ++i) {
      int m = mbase + i;
      int b_ = m / TT, t_ = m % TT;
      wq[(((size_t)b_ * NN + n) * TT + t_) * HH + h] = accs[sub][i];
    }
  }
#undef LD_SET
#undef WMMA_SET
}

// =====================================================================
// K1b: y[(b*T+s)*N + n] = x[b,s,:] . scaler[n,:]
// =====================================================================
__global__ void k1b_y(const float* __restrict__ X,
                      const float* __restrict__ scaler,
                      float* __restrict__ y) {
  int idx = blockIdx.x * blockDim.x + threadIdx.x;  // over MM*NN
  int n = idx & (NN - 1);
  int m = idx >> 4;
  const float* xr = X + (size_t)m * DD;
  const float* sr = scaler + (size_t)n * DD;
  float acc = 0.f;
  for (int d = 0; d < DD; ++d) acc += xr[d] * sr[d];
  y[idx] = acc;
}

// =====================================================================
// K2: one wave per (b,n,t): attn-row logits (K=16 dot), softmax,
// dot with y[b,:,n], sigmoid -> gate[(b*T+t)*N + n]
// =====================================================================
__global__ void k2_gate(const float* __restrict__ wq,
                        const float* __restrict__ y,
                        float* __restrict__ gate) {
  const int lane  = threadIdx.x & 31;
  const int gwave = blockIdx.x * (blockDim.x >> 5) + (threadIdx.x >> 5);
  const int t  = gwave % TT;
  const int bn = gwave / TT;          // b*NN + n
  const int n  = bn & (NN - 1);
  const int b  = bn >> 4;

  const float* qbase = wq + (size_t)bn * TT * HH;
  float q[HH];
#pragma unroll
  for (int h = 0; h < HH; ++h) q[h] = qbase[t * HH + h];

  float logit[3];
  float mx = -3.4e38f;
#pragma unroll
  for (int j = 0; j < 3; ++j) {
    int s = lane + 32 * j;
    float acc = -3.4e38f;
    if (s < TT) {
      acc = 0.f;
#pragma unroll
      for (int h = 0; h < HH; ++h) acc += q[h] * qbase[s * HH + h];
      mx = fmaxf(mx, acc);
    }
    logit[j] = acc;
  }
  for (int off = 16; off > 0; off >>= 1) mx = fmaxf(mx, __shfl_xor(mx, off, 32));

  float se = 0.f, wsum = 0.f;
#pragma unroll
  for (int j = 0; j < 3; ++j) {
    int s = lane + 32 * j;
    if (s < TT) {
      float p = __expf(logit[j] - mx);
      se += p;
      wsum += p * y[(size_t)(b * TT + s) * NN + n];
    }
  }
  for (int off = 16; off > 0; off >>= 1) {
    se   += __shfl_xor(se, off, 32);
    wsum += __shfl_xor(wsum, off, 32);
  }
  if (lane == 0) {
    float g = wsum / se;
    gate[(size_t)(b * TT + t) * NN + n] = 1.f / (1.f + __expf(-g));
  }
}

// =====================================================================
// K3: one 256-thread block per (b,t): argmax concept, recompute its
// softmax attn row, edit = maxv * (p . x[b]), down = edit @ lora_down,
// up_ws = 0.25 * down @ lora_up.   (deterministic serial reductions)
// =====================================================================
__global__ void k3_edit(const float* __restrict__ X,
                        const float* __restrict__ wq,
                        const float* __restrict__ gate,
                        const float* __restrict__ lora_down,
                        const float* __restrict__ lora_up,
                        float* __restrict__ up_ws) {
  __shared__ float p[TT];
  __shared__ float red[4 * 256];
  __shared__ float downS[4];
  __shared__ float mxS, seS, maxvS;
  __shared__ int nselS;

  const int bt = blockIdx.x;
  const int b = bt / TT, t = bt % TT;
  const int tid = threadIdx.x;

  if (tid == 0) {
    const float* g = gate + (size_t)bt * NN;
    float mv = g[0]; int mi = 0;
    for (int n = 1; n < NN; ++n)
      if (g[n] > mv) { mv = g[n]; mi = n; }   // first-max ties like jnp.argmax
    nselS = mi; maxvS = mv;
  }
  __syncthreads();
  const int n = nselS;
  const float* qbase = wq + (size_t)(b * NN + n) * TT * HH;

  if (tid < TT) {
    float acc = 0.f;
#pragma unroll
    for (int h = 0; h < HH; ++h) acc += qbase[t * HH + h] * qbase[tid * HH + h];
    p[tid] = acc;
  }
  __syncthreads();
  if (tid == 0) {
    float mx = p[0];
    for (int s = 1; s < TT; ++s) mx = fmaxf(mx, p[s]);
    float se = 0.f;
    for (int s = 0; s < TT; ++s) se += __expf(p[s] - mx);
    mxS = mx; seS = se;
  }
  __syncthreads();
  if (tid < TT) p[tid] = __expf(p[tid] - mxS) / seS * maxvS;  // fold max_val in
  __syncthreads();

  const float* xb = X + (size_t)b * TT * DD;
  float dpart[RR] = {0.f, 0.f, 0.f, 0.f};
#pragma unroll
  for (int rep = 0; rep < 3; ++rep) {
    int d = tid + rep * 256;                    // 768 = 3 * 256
    float e = 0.f;
    for (int s = 0; s < TT; ++s) e += p[s] * xb[(size_t)s * DD + d];
#pragma unroll
    for (int r = 0; r < RR; ++r)
      dpart[r] += e * lora_down[((size_t)n * DD + d) * RR + r];
  }
#pragma unroll
  for (int r = 0; r < RR; ++r) red[r * 256 + tid] = dpart[r];
  __syncthreads();
  if (tid < RR) {
    float s = 0.f;
    for (int i = 0; i < 256; ++i) s += red[tid * 256 + i];
    downS[tid] = s;
  }
  __syncthreads();

  const float* lu = lora_up + (size_t)n * RR * OO;
  float* upo = up_ws + (size_t)bt * OO;
  for (int o = tid; o < OO; o += 256) {
    float u = downS[0] * lu[o] + downS[1] * lu[OO + o] +
              downS[2] * lu[2 * OO + o] + downS[3] * lu[3 * OO + o];
    upo[o] = 0.25f * u;                          // MULTIPLIER * ALPHA / RANK
  }
}

// =====================================================================
// K4: out[m,o] = x[m,:] @ W_org[:,o] + b_org[o] + up_ws[m,o]
// 64x16 tile per wave, ping-pong double-buffered fragments.
// =====================================================================
__global__ void __attribute__((amdgpu_waves_per_eu(1)))
k4_out_wmma(const float* __restrict__ X,
            const float* __restrict__ W,
            const float* __restrict__ bias,
            const float* __restrict__ up_ws,
            float* __restrict__ out) {
  const int lane = threadIdx.x & 31;
  const int wave = threadIdx.x >> 5;
  const int tile = blockIdx.x * 4 + wave;       // MTILES*OTILES = 3080 exactly
  const int tn = tile % OTILES;
  const int tm = tile / OTILES;                 // 0..153
  const int lrow = lane & 15;
  const int kseg = lane >> 4;
  const int col = tn * 16 + (lane & 15);

  const float* xa0 = X + (size_t)(tm * MBLK +  0 + lrow) * DD + 2 * kseg;
  const float* xa1 = X + (size_t)(tm * MBLK + 16 + lrow) * DD + 2 * kseg;
  const float* xa2 = X + (size_t)(tm * MBLK + 32 + lrow) * DD + 2 * kseg;
  const float* xa3 = X + (size_t)(tm * MBLK + 48 + lrow) * DD + 2 * kseg;
  const float* wp  = W + (size_t)(2 * kseg) * OO + col;

  v8f acc0 = {}, acc1 = {}, acc2 = {}, acc3 = {};

  v2f bA, a0A, a1A, a2A, a3A;
  v2f bB, a0B, a1B, a2B, a3B;

#define LD_SET(kk, bv, f0, f1, f2, f3)                   \
  bv.x = wp[(size_t)(kk) * OO];                          \
  bv.y = wp[(size_t)(kk) * OO + OO];                     \
  f0 = *(const v2f*)(xa0 + (kk));                        \
  f1 = *(const v2f*)(xa1 + (kk));                        \
  f2 = *(const v2f*)(xa2 + (kk));                        \
  f3 = *(const v2f*)(xa3 + (kk));

#define WMMA_SET(bv, f0, f1, f2, f3)                                                              \
  acc0 = __builtin_amdgcn_wmma_f32_16x16x4_f32(false, f0, false, bv, (short)0, acc0, false, false); \
  acc1 = __builtin_amdgcn_wmma_f32_16x16x4_f32(false, f1, false, bv, (short)0, acc1, false, false); \
  acc2 = __builtin_amdgcn_wmma_f32_16x16x4_f32(false, f2, false, bv, (short)0, acc2, false, false); \
  acc3 = __builtin_amdgcn_wmma_f32_16x16x4_f32(false, f3, false, bv, (short)0, acc3, false, false);

  LD_SET(0, bA, a0A, a1A, a2A, a3A)
  int k = 0;
  for (; k < DD - 8; k += 8) {
    LD_SET(k + 4, bB, a0B, a1B, a2B, a3B)
    WMMA_SET(bA, a0A, a1A, a2A, a3A)
    LD_SET(k + 8, bA, a0A, a1A, a2A, a3A)
    WMMA_SET(bB, a0B, a1B, a2B, a3B)
  }
  LD_SET(k + 4, bB, a0B, a1B, a2B, a3B)
  WMMA_SET(bA, a0A, a1A, a2A, a3A)
  WMMA_SET(bB, a0B, a1B, a2B, a3B)

  const float bval = bias[col];
  v8f accs[4] = {acc0, acc1, acc2, acc3};
#pragma unroll
  for (int sub = 0; sub < 4; ++sub) {
    const int mbase = tm * MBLK + sub * 16 + (kseg ? 8 : 0);
#pragma unroll
    for (int i = 0; i < 8; ++i) {
      size_t idx = (size_t)(mbase + i) * OO + col;
      out[idx] = accs[sub][i] + bval + up_ws[idx];
    }
  }
#undef LD_SET
#undef WMMA_SET
}

// =====================================================================
extern "C" void kernel_launch(void* const* d_in, const int* in_sizes, int n_in,
                              void* d_out, int out_size, void* d_ws, size_t ws_size,
                              hipStream_t stream) {
  const float* x         = (const float*)d_in[0];  // [B,T,D]
  const float* w_qs      = (const float*)d_in[1];  // [N,D,H]
  const float* scaler    = (const float*)d_in[2];  // [N,D,1]
  const float* lora_down = (const float*)d_in[3];  // [N,D,R]
  const float* lora_up   = (const float*)d_in[4];  // [N,R,O]
  const float* W_org     = (const float*)d_in[5];  // [D,O]
  const float* b_org     = (const float*)d_in[6];  // [O]
  float* out = (float*)d_out;                      // [B,T,O]

  float* ws   = (float*)d_ws;
  float* wq   = ws;                                // B*N*T*H  = 2,523,136 floats
  float* y    = wq + (size_t)BB * NN * TT * HH;    // M*N      =   157,696
  float* gate = y + (size_t)MM * NN;               // M*N      =   157,696
  float* up   = gate + (size_t)MM * NN;            // M*O      = 3,153,920

  // K1: wq GEMM (WMMA, 64x16/wave) — 154*16 = 2464 wave-tiles, 4 waves/block
  k1_wq_wmma<<<(MTILES * NN) / 4, 128, 0, stream>>>(x, w_qs, wq);
  // K1b: y — MM*NN threads
  k1b_y<<<(MM * NN) / 256, 256, 0, stream>>>(x, scaler, y);
  // K2: gate — one wave per (b,n,t); 157,696 waves, 8 waves/block
  k2_gate<<<(BB * NN * TT) / 8, 256, 0, stream>>>(wq, y, gate);
  // K3: edit/LoRA — one block per (b,t)
  k3_edit<<<MM, 256, 0, stream>>>(x, wq, gate, lora_down, lora_up, up);
  // K4: output GEMM (WMMA, 64x16/wave) — 154*20 = 3080 wave-tiles, 4 waves/block
  k4_out_wmma<<<(MTILES * OTILES) / 4, 128, 0, stream>>>(x, W_org, b_org, up, out);
}